// Topological_Loss_43739946942654
// MI455X (gfx1250) — compile-verified
//
#include <hip/hip_runtime.h>
#include <hip/hip_bf16.h>
#include <cfloat>

#define NPTS 512
#define BATCH 32
#define TILE 128
#define KC 16
#define NCHUNK (NPTS / KC)
#define LDST 20   // LDS row stride in floats (16B-aligned rows, conflict-free fragment reads)

typedef float v2f __attribute__((ext_vector_type(2)));
typedef float v8f __attribute__((ext_vector_type(8)));

__device__ __forceinline__ float sigmoidf(float x) { return 1.0f / (1.0f + expf(-x)); }

__device__ __forceinline__ unsigned lds_off(const void* p) {
  // generic LDS pointer: addr[31:0] is the LDS byte address (CDNA5 aperture rule)
  return (unsigned)(uintptr_t)p;
}

// async 16B/lane copy global -> LDS, tracked on ASYNCcnt
__device__ __forceinline__ void async_copy16(unsigned lds_addr, const float* gaddr) {
  asm volatile("global_load_async_to_lds_b128 %0, %1, off"
               :: "v"(lds_addr), "v"(gaddr) : "memory");
}

// ---------------- Kernel 1: sigmoid materialization + row squared-norms + init ------
__global__ __launch_bounds__(256) void prep_kernel(
    const float* __restrict__ mo, const float* __restrict__ lab,
    float* __restrict__ P, float* __restrict__ sq,
    unsigned int* __restrict__ maxbits, float* __restrict__ loss)
{
  int bx = blockIdx.x;                 // 0 .. B*N-1
  int b = bx >> 9;
  int i = bx & (NPTS - 1);
  int tid = threadIdx.x;
  int lane = tid & 31, wid = tid >> 5;

  const float* mrow = mo + (size_t)b * NPTS * NPTS + (size_t)i * NPTS;
  const float* lrow = lab + (size_t)b * NPTS * NPTS + (size_t)i * NPTS;
  float* prow = P + (size_t)b * NPTS * NPTS + (size_t)i * NPTS;

  float sp = 0.f, sm = 0.f;
  for (int k = tid; k < NPTS; k += 256) {
    float s = sigmoidf(mrow[k]);
    prow[k] = s;
    sp += s * s;
    float m = lrow[k];
    sm += m * m;
  }
  for (int off = 16; off; off >>= 1) {
    sp += __shfl_xor(sp, off, 32);
    sm += __shfl_xor(sm, off, 32);
  }
  __shared__ float rp[8], rm[8];
  if (lane == 0) { rp[wid] = sp; rm[wid] = sm; }
  __syncthreads();
  if (tid == 0) {
    float a = 0.f, c = 0.f;
    for (int w = 0; w < 8; ++w) { a += rp[w]; c += rm[w]; }
    sq[(size_t)b * NPTS + i] = a;                     // cloud 0 (pred)
    sq[(size_t)(BATCH + b) * NPTS + i] = c;           // cloud 1 (mask)
  }
  if (bx < 2 * BATCH && tid == 0) { maxbits[bx] = 0u; loss[bx] = 0.f; }
}

// ---------------- Kernel 2: WMMA Gram GEMM (async double-buffered tiles) ------------
__global__ __launch_bounds__(256) void gram_dist_kernel(
    const float* __restrict__ P, const float* __restrict__ lab,
    float* __restrict__ Dx, float* __restrict__ Dz,
    const float* __restrict__ sq, unsigned int* __restrict__ maxbits)
{
  __shared__ float lA[2][TILE * LDST];
  __shared__ float lB[2][TILE * LDST];
  __shared__ float redmax[8];

  int tm = blockIdx.x;                 // output tile row  (N/TILE = 4)
  int tn = blockIdx.y;                 // output tile col
  int prob = blockIdx.z;               // c*32 + b
  int b = prob & 31;
  int c = prob >> 5;

  const float* X = (c == 0) ? (P + (size_t)b * NPTS * NPTS)
                            : (lab + (size_t)b * NPTS * NPTS);
  float* Dmat = ((c == 0) ? Dx : Dz) + (size_t)b * NPTS * NPTS;
  const float* sqv = sq + (size_t)prob * NPTS;

  int tid = threadIdx.x;
  int wid = tid >> 5;
  int lane = tid & 31;
  int wm = wid >> 2;                   // 0..1  -> 64-row strip
  int wn = wid & 3;                    // 0..3  -> 32-col strip
  int lo = lane & 15;
  int hi = lane >> 4;

  int rowA = tm * TILE;
  int rowB = tn * TILE;

  // per-thread async-staging geometry: two 16B slots per tile per chunk
  int idx0 = tid, idx1 = tid + 256;
  int r0 = idx0 >> 2, cg0 = idx0 & 3;
  int r1 = idx1 >> 2, cg1 = idx1 & 3;
  unsigned lAb[2] = { lds_off(&lA[0][0]), lds_off(&lA[1][0]) };
  unsigned lBb[2] = { lds_off(&lB[0][0]), lds_off(&lB[1][0]) };
  unsigned o0 = (unsigned)((r0 * LDST + cg0 * 4) * sizeof(float));
  unsigned o1 = (unsigned)((r1 * LDST + cg1 * 4) * sizeof(float));

  v8f acc[4][2];
  for (int i = 0; i < 4; ++i)
    for (int j = 0; j < 2; ++j)
      for (int e = 0; e < 8; ++e) acc[i][j][e] = 0.f;

  // prologue: stage chunk 0 into buffer 0  (4 async ops per thread -> ASYNCcnt += 4)
  async_copy16(lAb[0] + o0, X + (size_t)(rowA + r0) * NPTS + cg0 * 4);
  async_copy16(lBb[0] + o0, X + (size_t)(rowB + r0) * NPTS + cg0 * 4);
  async_copy16(lAb[0] + o1, X + (size_t)(rowA + r1) * NPTS + cg1 * 4);
  async_copy16(lBb[0] + o1, X + (size_t)(rowB + r1) * NPTS + cg1 * 4);

  for (int kc = 0; kc < NCHUNK; ++kc) {
    int cur = kc & 1;
    if (kc + 1 < NCHUNK) {
      int nb = cur ^ 1;
      int k0 = (kc + 1) * KC;
      async_copy16(lAb[nb] + o0, X + (size_t)(rowA + r0) * NPTS + k0 + cg0 * 4);
      async_copy16(lBb[nb] + o0, X + (size_t)(rowB + r0) * NPTS + k0 + cg0 * 4);
      async_copy16(lAb[nb] + o1, X + (size_t)(rowA + r1) * NPTS + k0 + cg1 * 4);
      async_copy16(lBb[nb] + o1, X + (size_t)(rowB + r1) * NPTS + k0 + cg1 * 4);
      // retire the 4 older ops (current chunk); in-order completion per wave
      asm volatile("s_wait_asynccnt 0x4" ::: "memory");
    } else {
      asm volatile("s_wait_asynccnt 0x0" ::: "memory");
    }
    __syncthreads();

    for (int kk = 0; kk < KC; kk += 4) {
      v2f afr[4], bfr[2];
      for (int i = 0; i < 4; ++i) {
        int m = wm * 64 + i * 16 + lo;
        afr[i] = *(const v2f*)&lA[cur][m * LDST + kk + 2 * hi];
      }
      for (int j = 0; j < 2; ++j) {
        int nn = wn * 32 + j * 16 + lo;
        bfr[j] = *(const v2f*)&lB[cur][nn * LDST + kk + 2 * hi];
      }
      for (int i = 0; i < 4; ++i)
        for (int j = 0; j < 2; ++j)
          acc[i][j] = __builtin_amdgcn_wmma_f32_16x16x4_f32(
              false, afr[i], false, bfr[j], (short)0, acc[i][j], false, false);
    }
    __syncthreads();
  }

  // epilogue: d = sqrt(max(sq_i + sq_j - 2g, 0)); track tile max
  float localmax = 0.f;
  for (int i = 0; i < 4; ++i)
    for (int j = 0; j < 2; ++j)
      for (int v = 0; v < 8; ++v) {
        int r = rowA + wm * 64 + i * 16 + v + 8 * hi;
        int cc = rowB + wn * 32 + j * 16 + lo;
        float g = acc[i][j][v];
        float d2 = sqv[r] + sqv[cc] - 2.0f * g;
        float d = sqrtf(fmaxf(d2, 0.0f));
        Dmat[(size_t)r * NPTS + cc] = d;
        localmax = fmaxf(localmax, d);
      }
  for (int off = 16; off; off >>= 1)
    localmax = fmaxf(localmax, __shfl_xor(localmax, off, 32));
  if (lane == 0) redmax[wid] = localmax;
  __syncthreads();
  if (wid == 0) {
    float m = (lane < 8) ? redmax[lane] : 0.f;
    for (int off = 4; off; off >>= 1) m = fmaxf(m, __shfl_xor(m, off, 32));
    if (lane == 0) atomicMax(&maxbits[prob], __float_as_uint(m));
  }
}

// ---------------- Kernel 3: Prim's MST per (b, cloud) + signature-loss accumulation --
__global__ __launch_bounds__(512) void mst_kernel(
    const float* __restrict__ Dx, const float* __restrict__ Dz,
    const unsigned int* __restrict__ maxbits, float* __restrict__ loss)
{
  int prob = blockIdx.x;               // c*32 + b
  int b = prob & 31;
  int c = prob >> 5;
  const float* D = ((c == 0) ? Dx : Dz) + (size_t)b * NPTS * NPTS;
  const float* DxB = Dx + (size_t)b * NPTS * NPTS;
  const float* DzB = Dz + (size_t)b * NPTS * NPTS;
  float invMx = 1.0f / fmaxf(__uint_as_float(maxbits[b]), 1e-12f);
  float invMz = 1.0f / fmaxf(__uint_as_float(maxbits[BATCH + b]), 1e-12f);

  int tid = threadIdx.x;               // one node per thread
  int lane = tid & 31, wid = tid >> 5;

  __shared__ float md[NPTS];
  __shared__ int mf[NPTS];
  __shared__ unsigned char intree[NPTS];
  __shared__ unsigned long long red[16];
  __shared__ unsigned long long bcastS;

  md[tid] = (tid == 0) ? FLT_MAX : D[tid];   // row 0 seeds frontier
  mf[tid] = 0;
  intree[tid] = (tid == 0) ? 1 : 0;
  float acc = 0.f;
  __syncthreads();

  for (int step = 0; step < NPTS - 1; ++step) {
    // argmin over frontier: pack (f32 bits, index) into u64; distances >= 0 so bits are monotone
    float mv = intree[tid] ? FLT_MAX : md[tid];
    unsigned long long key =
        ((unsigned long long)__float_as_uint(mv) << 32) | (unsigned int)tid;
    for (int off = 16; off; off >>= 1) {
      unsigned long long o = __shfl_xor(key, off, 32);
      key = (o < key) ? o : key;
    }
    if (lane == 0) red[wid] = key;
    __syncthreads();
    if (wid == 0) {
      unsigned long long k2 = (lane < 16) ? red[lane] : ~0ULL;
      for (int off = 8; off; off >>= 1) {
        unsigned long long o = __shfl_xor(k2, off, 32);
        k2 = (o < k2) ? o : k2;
      }
      if (lane == 0) bcastS = k2;
    }
    __syncthreads();
    int j = (int)(unsigned int)bcastS;

    if (tid == 0) {                    // record MST edge (mf[j], j), accumulate loss term
      int i = mf[j];
      float ex = DxB[(size_t)i * NPTS + j] * invMx;
      float ez = DzB[(size_t)i * NPTS + j] * invMz;
      float e = ex - ez;
      acc += e * e;
    }
    if (tid == j) intree[tid] = 1;     // own-slot write; no race with own read below
    float d = D[(size_t)j * NPTS + tid];
    if (!intree[tid] && d < md[tid]) { md[tid] = d; mf[tid] = j; }
    __syncthreads();
  }
  if (tid == 0) loss[prob] = acc;
}

// ---------------- Kernel 4: final reduction ------------------------------------------
__global__ void finalize_kernel(const float* __restrict__ loss, float* __restrict__ out) {
  __shared__ float s[2 * BATCH];
  int t = threadIdx.x;                 // 64 threads
  s[t] = loss[t];
  __syncthreads();
  if (t == 0) {
    float tot = 0.f;
    for (int i = 0; i < 2 * BATCH; ++i) tot += s[i];
    out[0] = tot * (1.0f / (float)BATCH);   // LAM = 1
  }
}

extern "C" void kernel_launch(void* const* d_in, const int* in_sizes, int n_in,
                              void* d_out, int out_size, void* d_ws, size_t ws_size,
                              hipStream_t stream) {
  const float* mo  = (const float*)d_in[0];   // model_output (B,1,H,W) f32
  const float* lab = (const float*)d_in[1];   // labels       (B,1,H,W) f32

  char* ws = (char*)d_ws;
  size_t nmat = (size_t)BATCH * NPTS * NPTS;           // 8388608 floats per matrix set
  float*        Dx      = (float*)ws;                  // 32 MB
  float*        Dz      = (float*)(ws + nmat * 4);     // 32 MB
  float*        Pm      = (float*)(ws + 2 * nmat * 4); // 32 MB sigmoid(preds)
  float*        sq      = (float*)(ws + 3 * nmat * 4);               // 2*32*512 f32
  unsigned int* maxbits = (unsigned int*)(ws + 3 * nmat * 4 + 2 * BATCH * NPTS * 4);
  float*        loss    = (float*)((char*)maxbits + 2 * BATCH * 4);

  prep_kernel<<<dim3(BATCH * NPTS), 256, 0, stream>>>(mo, lab, Pm, sq, maxbits, loss);
  gram_dist_kernel<<<dim3(NPTS / TILE, NPTS / TILE, 2 * BATCH), 256, 0, stream>>>(
      Pm, lab, Dx, Dz, sq, maxbits);
  mst_kernel<<<dim3(2 * BATCH), 512, 0, stream>>>(Dx, Dz, maxbits, loss);
  finalize_kernel<<<1, 2 * BATCH, 0, stream>>>(loss, (float*)d_out);
}